// ReservoirComputerBase_43568148250942
// MI455X (gfx1250) — compile-verified
//
#include <hip/hip_runtime.h>
#include <math.h>

// ESN teacher-forced rollout for MI455X (gfx1250, wave32).
//   SEQ_LEN=8192, IN_DIM=64, RES_DIM=2048
// Phase 1: proj = in_seq @ W_in^T + bias  via V_WMMA_F32_16X16X4_F32 -> d_out
// Phase 2: persistent scan, W_res held in VGPRs, grid barrier per step,
//          states overwrite d_out rows in place.

#define SEQ_LEN 8192
#define IN_DIM  64
#define RES_DIM 2048

#define SCAN_NBLK 128          // persistent blocks
#define SCAN_BLK  512          // 16 waves/block -> 2048 waves = one per reservoir row

typedef float v2f __attribute__((ext_vector_type(2)));
typedef float v8f __attribute__((ext_vector_type(8)));

// ---------------------------------------------------------------------------
// Phase 1: (8192x64)x(64x2048) GEMM, one wave per 16x16 output tile.
// K=64 -> 16 chained v_wmma_f32_16x16x4_f32 per tile.
// A frag (16x4 f32, 2 VGPR): lane l, vgpr v -> M=l%16, K=v+2*(l/16)
// B frag (4x16 f32, 2 VGPR): lane l, vgpr v -> N=l%16, K=v+2*(l/16)  (mirrored)
// C/D (16x16 f32, 8 VGPR):   vgpr g, lanes0-15 -> M=g,N=l ; lanes16-31 -> M=g+8
// ---------------------------------------------------------------------------
__global__ void __launch_bounds__(256)
esn_proj_gemm(const float* __restrict__ in_seq,
              const float* __restrict__ W_in,
              const float* __restrict__ bias,
              float* __restrict__ out)
{
    const int wave_id = (blockIdx.x * blockDim.x + threadIdx.x) >> 5;
    const int lane    = threadIdx.x & 31;
    const int lo      = lane & 15;
    const int hi      = lane >> 4;               // 0 or 1

    const int tiles_n = RES_DIM / 16;            // 128
    const int tm = wave_id / tiles_n;            // 0..511  (seq tile)
    const int tn = wave_id % tiles_n;            // 0..127  (reservoir tile)

    // accumulator pre-loaded with bias (bias[n] replicated down M)
    const float b = bias[tn * 16 + lo];
    v8f c;
#pragma unroll
    for (int i = 0; i < 8; ++i) c[i] = b;

    const float* arow = in_seq + (size_t)(tm * 16 + lo) * IN_DIM;  // A: row M=lo
    const float* brow = W_in   + (size_t)(tn * 16 + lo) * IN_DIM;  // B^T row N=lo

#pragma unroll
    for (int k = 0; k < IN_DIM; k += 4) {
        v2f a  = *(const v2f*)(arow + k + 2 * hi);   // K = k+2hi, k+2hi+1
        v2f bb = *(const v2f*)(brow + k + 2 * hi);
        c = __builtin_amdgcn_wmma_f32_16x16x4_f32(
                /*neg_a=*/false, a, /*neg_b=*/false, bb,
                /*c_mod=*/(short)0, c, /*reuse_a=*/false, /*reuse_b=*/false);
    }

    float* orow = out + (size_t)(tm * 16) * RES_DIM + tn * 16;
#pragma unroll
    for (int g = 0; g < 8; ++g)
        orow[(size_t)(g + 8 * hi) * RES_DIM + lo] = c[g];
}

// ---------------------------------------------------------------------------
// init: ping-pong state buffers + grid-barrier words (graph-safe, per launch)
// ---------------------------------------------------------------------------
__global__ void esn_init(const float* __restrict__ res_state,
                         float* __restrict__ st0, float* __restrict__ st1,
                         unsigned* __restrict__ cnt, unsigned* __restrict__ gen)
{
    const int i = blockIdx.x * blockDim.x + threadIdx.x;
    if (i < RES_DIM) { st0[i] = res_state[i]; st1[i] = res_state[i]; }
    if (i == 0)      { *cnt = 0u; *gen = 0u; }
}

// ---------------------------------------------------------------------------
// Phase 2: persistent scan. wave w of block b owns row r = b*16+w.
// Lane l holds W_res[r][j*64 + 2l .. 2l+1] for j=0..31 in 64 VGPRs.
// State broadcast global->LDS each step; b64 LDS reads are bank-conflict-free.
// ---------------------------------------------------------------------------
__global__ void __launch_bounds__(SCAN_BLK, 1)
esn_scan(const float* __restrict__ W_res,
         float* __restrict__ seq,            // d_out: proj+bias in, states out
         float* __restrict__ st0,
         float* __restrict__ st1,
         unsigned* __restrict__ cnt,
         unsigned* __restrict__ gen)
{
    __shared__ float s[RES_DIM];

    const int tid  = threadIdx.x;
    const int wave = tid >> 5;
    const int lane = tid & 31;
    const int row  = blockIdx.x * (SCAN_BLK / 32) + wave;   // 0..2047

    // --- preload this row's weights into registers (16 MB grid-wide) ---
    float w[64];
    {
        const float* wr = W_res + (size_t)row * RES_DIM + 2 * lane;
#pragma unroll
        for (int j = 0; j < 32; ++j) {
            v2f t = *(const v2f*)(wr + j * 64);
            w[2 * j]     = t[0];
            w[2 * j + 1] = t[1];
        }
    }

    unsigned local_gen = 0;

    for (int t = 0; t < SEQ_LEN; ++t) {
        const float* rbuf = (t & 1) ? st0 : st1;   // t=0 reads st1 (init state)
        float*       wbuf = (t & 1) ? st1 : st0;

        // broadcast state into LDS (L2-resident, 8 KB)
        for (int i = tid; i < RES_DIM; i += SCAN_BLK) s[i] = rbuf[i];
        __syncthreads();

        // 64 FMAs/lane dot-product slice
        float acc = 0.0f;
#pragma unroll
        for (int j = 0; j < 32; ++j) {
            v2f ls = *(const v2f*)&s[j * 64 + 2 * lane];
            acc = fmaf(w[2 * j],     ls[0], acc);
            acc = fmaf(w[2 * j + 1], ls[1], acc);
        }
        // wave32 reduction
#pragma unroll
        for (int off = 16; off > 0; off >>= 1)
            acc += __shfl_xor(acc, off, 32);

        if (lane == 0) {
            const float v = tanhf(acc + seq[(size_t)t * RES_DIM + row]);
            seq[(size_t)t * RES_DIM + row] = v;    // emit state (overwrites proj)
            wbuf[row] = v;                         // feed next step
        }

        // ---- grid-wide barrier (sense via monotone generation counter) ----
        __threadfence();                 // make state/seq writes device-visible
        __syncthreads();
        if (tid == 0) {
            const unsigned target = local_gen + 1u;
            const unsigned prev = atomicAdd(cnt, 1u);
            if (prev == SCAN_NBLK - 1u) {
                atomicExch(cnt, 0u);
                __threadfence();
                atomicAdd(gen, 1u);      // release
            } else {
                while (atomicAdd(gen, 0u) < target) {
                    __builtin_amdgcn_s_sleep(1);
                }
            }
        }
        local_gen += 1u;
        __syncthreads();                 // also guards LDS reuse next iteration
        __threadfence();
    }
}

extern "C" void kernel_launch(void* const* d_in, const int* in_sizes, int n_in,
                              void* d_out, int out_size, void* d_ws, size_t ws_size,
                              hipStream_t stream)
{
    const float* in_seq    = (const float*)d_in[0];  // 8192*64
    const float* res_state = (const float*)d_in[1];  // 2048
    const float* W_in      = (const float*)d_in[2];  // 2048*64
    const float* W_res     = (const float*)d_in[3];  // 2048*2048
    const float* bias      = (const float*)d_in[4];  // 2048

    float* seq_out = (float*)d_out;                  // 8192*2048

    // workspace: two 2048-f32 state buffers + 2 barrier words
    float*    st0 = (float*)d_ws;
    float*    st1 = st0 + RES_DIM;
    unsigned* cnt = (unsigned*)(st1 + RES_DIM);
    unsigned* gen = cnt + 1;

    // init ping-pong state + barrier
    esn_init<<<(RES_DIM + 255) / 256, 256, 0, stream>>>(res_state, st0, st1, cnt, gen);

    // proj+bias GEMM via WMMA: (8192/16)*(2048/16)=65536 tiles, 1 wave each
    {
        const int total_waves   = (SEQ_LEN / 16) * (RES_DIM / 16);
        const int waves_per_blk = 256 / 32;
        const int blocks        = total_waves / waves_per_blk;   // 8192
        esn_proj_gemm<<<blocks, 256, 0, stream>>>(in_seq, W_in, bias, seq_out);
    }

    // persistent sequential rollout
    esn_scan<<<SCAN_NBLK, SCAN_BLK, 0, stream>>>(W_res, seq_out, st0, st1, cnt, gen);
}